// SlotPredictor_55808805044465
// MI455X (gfx1250) — compile-verified
//
#include <hip/hip_runtime.h>
#include <hip/hip_bf16.h>

typedef __attribute__((ext_vector_type(2))) float v2f;
typedef __attribute__((ext_vector_type(8))) float v8f;

#define BSZ 128
#define SEQ 512
#define HID 512
#define NTAG 72
#define NTP 80   // padded N (5 tiles of 16)

// ---------------------------------------------------------------------------
// Kernel 1: feats = inputs @ W + b   (65536x512 @ 512x72), f32 WMMA 16x16x4.
// Block = 256 threads (8 waves); each wave owns a 16-row M strip -> 128 rows
// per block, 512 blocks. W is staged in LDS in 64-K chunks, zero-padded to 80
// columns so B fragments need no bounds checks.
// ---------------------------------------------------------------------------
__global__ __launch_bounds__(256) void feats_gemm(
    const float* __restrict__ x, const float* __restrict__ W,
    const float* __restrict__ bias, float* __restrict__ feats) {
  __shared__ float lw[64 * NTP];  // 20 KB

  const int tid  = threadIdx.x;
  const int wave = tid >> 5;
  const int lane = tid & 31;
  const int lo   = lane & 15;     // M (for A) / N (for B) within tile
  const int hi   = lane >> 4;     // K half selector
  const int m0   = (blockIdx.x * 8 + wave) << 4;

  v8f acc[5];
  const v8f vzero = {0.f, 0.f, 0.f, 0.f, 0.f, 0.f, 0.f, 0.f};
#pragma unroll
  for (int nt = 0; nt < 5; ++nt) acc[nt] = vzero;

  // lane's A stream: row m0+lo, K offset hi*2 (+ {0,1} via 8B load)
  const float* arow = x + (size_t)(m0 + lo) * HID + hi * 2;

  for (int kc = 0; kc < HID / 64; ++kc) {
    const int k0 = kc * 64;
    __syncthreads();
    // cooperative, fully-uniform stage of W[k0:k0+64, 0:72] -> lw[64][80]
    for (int e = tid; e < 64 * NTP; e += 256) {
      int r = e / NTP;
      int c = e - r * NTP;
      lw[e] = (c < NTAG) ? W[(k0 + r) * NTAG + c] : 0.f;
    }
    __syncthreads();

    // hint the next A chunk into the cache hierarchy (global_prefetch_b8)
    __builtin_prefetch(arow + k0 + 64, 0, 1);

#pragma unroll 4
    for (int kk = 0; kk < 16; ++kk) {
      const int k = kk * 4;
      const float2 ax = *(const float2*)(arow + k0 + k);
      v2f av;
      av.x = ax.x;
      av.y = ax.y;
      const int krow = k + hi * 2;
#pragma unroll
      for (int nt = 0; nt < 5; ++nt) {
        v2f bv;
        bv.x = lw[krow * NTP + nt * 16 + lo];
        bv.y = lw[(krow + 1) * NTP + nt * 16 + lo];
        acc[nt] = __builtin_amdgcn_wmma_f32_16x16x4_f32(
            false, av, false, bv, (short)0, acc[nt], false, false);
      }
    }
  }

  // epilogue: D layout = (M = r + 8*hi, N = lo); add bias, drop padded cols
#pragma unroll
  for (int nt = 0; nt < 5; ++nt) {
    const int col = nt * 16 + lo;
    if (col < NTAG) {
      const float bb = bias[col];
#pragma unroll
      for (int r = 0; r < 8; ++r) {
        const int row = m0 + r + 8 * hi;
        feats[(size_t)row * NTAG + col] = acc[nt][r] + bb;
      }
    }
  }
}

// ---------------------------------------------------------------------------
// Kernel 2: CRF forward scan + sequence score, one block per batch element.
// 288 threads = 4 i-segments x 72 j-states (9 wave32s). expT = exp(T) is
// precomputed once into LDS, so each step is: wave0 max-reduce, 72 exps,
// a 4-way-split 72x72 mat-vec (18 FMAs/thread), 72 logs.
// ---------------------------------------------------------------------------
__global__ __launch_bounds__(288) void crf_scan(
    const float* __restrict__ feats, const float* __restrict__ T,
    const float* __restrict__ startT, const float* __restrict__ stopT,
    const int* __restrict__ tags, float* __restrict__ partials) {
  __shared__ float expT[NTAG * NTAG];  // 20.7 KB
  __shared__ float aL[NTAG];
  __shared__ float eL[NTAG];
  __shared__ float ps[4 * NTAG];
  __shared__ float red[288];
  __shared__ float sAmax, sSeq;

  const int tid = threadIdx.x;
  const int b   = blockIdx.x;
  const int seg = tid / NTAG;       // 0..3
  const int j   = tid - seg * NTAG; // 0..71
  const float* fb = feats + (size_t)b * SEQ * NTAG;
  const int*   tg = tags + b * SEQ;

  // exp(T) once per block
  for (int e = tid; e < NTAG * NTAG; e += 288) expT[e] = __expf(T[e]);

  // gold-path score partials (emission + transition gathers)
  float sc = 0.f;
  for (int t = tid; t < SEQ; t += 288) {
    const int tt = tg[t];
    sc += fb[t * NTAG + tt];
    if (t < SEQ - 1) sc += T[tt * NTAG + tg[t + 1]];
  }
  red[tid] = sc;

  if (tid < NTAG) aL[tid] = fb[tid] + startT[tid];  // a0
  __syncthreads();
  if (tid == 0) {
    float s = 0.f;
    for (int i = 0; i < 288; ++i) s += red[i];
    sSeq = s + startT[tg[0]] + stopT[tg[SEQ - 1]];
  }
  __syncthreads();

  for (int t = 1; t < SEQ; ++t) {
    // wave 0: amax = max_i a[i]  (72 = 32 + 32 + 8)
    if (tid < 32) {
      float m = fmaxf(aL[tid], aL[tid + 32]);
      if (tid < 8) m = fmaxf(m, aL[tid + 64]);
#pragma unroll
      for (int off = 16; off; off >>= 1) m = fmaxf(m, __shfl_xor(m, off, 32));
      if (tid == 0) sAmax = m;
    }
    __syncthreads();
    const float amax = sAmax;
    if (seg == 0) eL[j] = __expf(aL[j] - amax);
    __syncthreads();
    {
      float s = 0.f;
      const float* er = expT + seg * 18 * NTAG + j;
#pragma unroll
      for (int ii = 0; ii < 18; ++ii) s = fmaf(eL[seg * 18 + ii], er[ii * NTAG], s);
      ps[seg * NTAG + j] = s;
    }
    __syncthreads();
    if (seg == 0) {
      const float tot = ps[j] + ps[NTAG + j] + ps[2 * NTAG + j] + ps[3 * NTAG + j];
      aL[j] = amax + __logf(tot) + fb[t * NTAG + j];
    }
    __syncthreads();
  }

  // log_z = logsumexp(a + stop); wave 0 only
  if (tid < 32) {
    const float v1 = aL[tid] + stopT[tid];
    const float v2 = aL[tid + 32] + stopT[tid + 32];
    const float v3 = (tid < 8) ? (aL[tid + 64] + stopT[tid + 64]) : -3.4e38f;
    float m = fmaxf(fmaxf(v1, v2), v3);
#pragma unroll
    for (int off = 16; off; off >>= 1) m = fmaxf(m, __shfl_xor(m, off, 32));
    float s = __expf(v1 - m) + __expf(v2 - m) + ((tid < 8) ? __expf(v3 - m) : 0.f);
#pragma unroll
    for (int off = 16; off; off >>= 1) s += __shfl_xor(s, off, 32);
    if (tid == 0) partials[b] = sSeq - (m + __logf(s));
  }
}

// ---------------------------------------------------------------------------
// Kernel 3: deterministic mean over the 128 per-batch results.
// ---------------------------------------------------------------------------
__global__ void crf_finalize(const float* __restrict__ partials,
                             float* __restrict__ out) {
  const int lane = threadIdx.x;
  float v = partials[lane] + partials[lane + 32] + partials[lane + 64] +
            partials[lane + 96];
#pragma unroll
  for (int off = 16; off; off >>= 1) v += __shfl_xor(v, off, 32);
  if (lane == 0) out[0] = -v / (float)BSZ;
}

extern "C" void kernel_launch(void* const* d_in, const int* in_sizes, int n_in,
                              void* d_out, int out_size, void* d_ws,
                              size_t ws_size, hipStream_t stream) {
  (void)in_sizes; (void)n_in; (void)out_size; (void)ws_size;
  const float* x      = (const float*)d_in[0];
  const float* W      = (const float*)d_in[1];
  const float* bias   = (const float*)d_in[2];
  const float* T      = (const float*)d_in[3];
  const float* startT = (const float*)d_in[4];
  const float* stopT  = (const float*)d_in[5];
  const int*   tags   = (const int*)d_in[6];
  float* out = (float*)d_out;

  float* feats    = (float*)d_ws;                       // 65536*72 f32 (~18.9 MB)
  float* partials = feats + (size_t)BSZ * SEQ * NTAG;   // 128 f32

  feats_gemm<<<(BSZ * SEQ) / 128, 256, 0, stream>>>(x, W, bias, feats);
  crf_scan<<<BSZ, 288, 0, stream>>>(feats, T, startT, stopT, tags, partials);
  crf_finalize<<<1, 32, 0, stream>>>(partials, out);
}